// Listnet_For_Gauss_55413668053158
// MI455X (gfx1250) — compile-verified
//
#include <hip/hip_runtime.h>
#include <math.h>

#define SEG_THREADS 256
#define NWAVES (SEG_THREADS / 32)

typedef __attribute__((ext_vector_type(2))) float v2f;
typedef __attribute__((ext_vector_type(8))) float v8f;

__device__ __forceinline__ float wave_sum(float v) {
#pragma unroll
  for (int off = 16; off; off >>= 1) v += __shfl_xor(v, off, 32);
  return v;
}
__device__ __forceinline__ float wave_max(float v) {
#pragma unroll
  for (int off = 16; off; off >>= 1) v = fmaxf(v, __shfl_xor(v, off, 32));
  return v;
}

// One workgroup per segment (seg_len == 512, 256 threads, 2 elements/thread).
// Single pass over HBM: each element loaded once (float2), all reductions in
// registers/LDS. Memory-bound: 12 B/element, ~100 MB total -> ~4.3 us floor.
__global__ __launch_bounds__(SEG_THREADS) void seg_loss_kernel(
    const float* __restrict__ mean, const float* __restrict__ var,
    const int* __restrict__ scope, const float* __restrict__ targets,
    float* __restrict__ seg_loss, int seg_len) {
  const int seg = blockIdx.x;
  const long base = (long)seg * (long)seg_len;
  const int tid = threadIdx.x;
  const int lane = tid & 31, wid = tid >> 5;
  __shared__ float rA[NWAVES], rT[NWAVES], rSA[NWAVES], rST[NWAVES], rD[NWAVES];

  const int i0 = tid * 2;
  const bool v0 = i0 < seg_len, v1 = (i0 + 1) < seg_len;
  float x0 = 0.f, x1 = 0.f, y0 = 0.f, y1 = 0.f, t0 = 0.f, t1 = 0.f;
  if (v1) {  // the common (uniform-512) path: one b64 load per array
    v2f mx = *(const v2f*)(mean + base + i0);
    v2f vy = *(const v2f*)(var + base + i0);
    v2f tt = *(const v2f*)(targets + base + i0);
    x0 = mx.x; x1 = mx.y; y0 = vy.x; y1 = vy.y; t0 = tt.x; t1 = tt.y;
  } else if (v0) {
    x0 = mean[base + i0]; y0 = var[base + i0]; t0 = targets[base + i0];
  }

  float a0 = fmaf(0.5f, y0, x0), a1 = fmaf(0.5f, y1, x1);
  const float NEGBIG = -3.402823466e38f;
  float am = fmaxf(v0 ? a0 : NEGBIG, v1 ? a1 : NEGBIG);
  float tm = fmaxf(v0 ? t0 : NEGBIG, v1 ? t1 : NEGBIG);
  am = wave_max(am); tm = wave_max(tm);
  if (lane == 0) { rA[wid] = am; rT[wid] = tm; }
  __syncthreads();
  float m_a = rA[0], m_t = rT[0];
#pragma unroll
  for (int w = 1; w < NWAVES; ++w) { m_a = fmaxf(m_a, rA[w]); m_t = fmaxf(m_t, rT[w]); }
  if (!isfinite(m_a)) m_a = 0.f;
  if (!isfinite(m_t)) m_t = 0.f;

  float sa = (v0 ? __expf(a0 - m_a) : 0.f) + (v1 ? __expf(a1 - m_a) : 0.f);
  float st = (v0 ? __expf(t0 - m_t) : 0.f) + (v1 ? __expf(t1 - m_t) : 0.f);
  sa = wave_sum(sa); st = wave_sum(st);
  if (lane == 0) { rSA[wid] = sa; rST[wid] = st; }
  __syncthreads();
  float Sa = 0.f, St = 0.f;
#pragma unroll
  for (int w = 0; w < NWAVES; ++w) { Sa += rSA[w]; St += rST[w]; }
  const float logS = m_a + __logf(Sa);   // logsumexp(a)
  const float logZ = m_t + __logf(St);   // logsumexp(targets)

  float d = 0.f;
  if (v0) d += __expf(t0 - logZ) * (x0 - 0.5f * y0 - logS);
  if (v1) d += __expf(t1 - logZ) * (x1 - 0.5f * y1 - logS);
  d = wave_sum(d);
  if (lane == 0) rD[wid] = d;
  __syncthreads();
  if (tid == 0) {
    float tot = 0.f;
#pragma unroll
    for (int w = 0; w < NWAVES; ++w) tot += rD[w];
    seg_loss[seg] = -tot / (float)scope[seg];
  }
}

// Deterministic final mean over per-segment losses using the WMMA pipe:
// D = A(16x4 f32) * B(4x16 ones) + C sums 64 f32 values per instruction
// with native f32 accumulation in C. Layout permutation of A is irrelevant
// because we reduce D to a single total. One workgroup, 8 waves, EXEC all-1s
// throughout the WMMA loop (uniform trip count, all 32 lanes active).
__global__ __launch_bounds__(SEG_THREADS) void final_mean_kernel(
    const float* __restrict__ seg_loss, float* __restrict__ out, int num_seg) {
  const int tid = threadIdx.x;
  const int lane = tid & 31, wid = tid >> 5;
  const int full = (num_seg / (NWAVES * 64)) * (NWAVES * 64);
  const int per_wave = full / NWAVES;           // multiple of 64
  const int wbase = wid * per_wave;

  v8f c = {0.f, 0.f, 0.f, 0.f, 0.f, 0.f, 0.f, 0.f};
  v2f ones; ones.x = 1.f; ones.y = 1.f;
  for (int it = 0; it < per_wave; it += 64) {
    v2f a = *(const v2f*)(seg_loss + wbase + it + lane * 2);
    // (neg_a, A, neg_b, B, c_mod, C, reuse_a, reuse_b)
    c = __builtin_amdgcn_wmma_f32_16x16x4_f32(
        /*neg_a=*/false, a, /*neg_b=*/false, ones,
        /*c_mod=*/(short)0, c, /*reuse_a=*/false, /*reuse_b=*/false);
  }
  // lanes 0-15 hold D[M=0..7, n], lanes 16-31 hold D[M=8..15, n]
  float s = c[0] + c[1] + c[2] + c[3] + c[4] + c[5] + c[6] + c[7];
  s += __shfl_xor(s, 16, 32);  // fold the two M-halves -> wave total

  // scalar tail (empty for num_seg = 16384)
  float tp = 0.f;
  for (int i = full + tid; i < num_seg; i += SEG_THREADS) tp += seg_loss[i];
  s += wave_sum(tp);

  __shared__ float wsum[NWAVES];
  if (lane == 0) wsum[wid] = s;
  __syncthreads();
  if (tid == 0) {
    float tot = 0.f;
#pragma unroll
    for (int w = 0; w < NWAVES; ++w) tot += wsum[w];
    out[0] = tot / (float)num_seg;
  }
}

extern "C" void kernel_launch(void* const* d_in, const int* in_sizes, int n_in,
                              void* d_out, int out_size, void* d_ws, size_t ws_size,
                              hipStream_t stream) {
  const float* mean    = (const float*)d_in[0];
  const float* var     = (const float*)d_in[1];
  const int*   scope   = (const int*)d_in[2];
  const float* targets = (const float*)d_in[3];
  float* out = (float*)d_out;
  float* seg_loss = (float*)d_ws;   // num_seg floats of scratch

  const int num_seg = in_sizes[2];
  const int seg_len = in_sizes[0] / num_seg;   // 512

  seg_loss_kernel<<<num_seg, SEG_THREADS, 0, stream>>>(
      mean, var, scope, targets, seg_loss, seg_len);
  final_mean_kernel<<<1, SEG_THREADS, 0, stream>>>(seg_loss, out, num_seg);
}